// Router_1580547968683
// MI455X (gfx1250) — compile-verified
//
#include <hip/hip_runtime.h>

// CapsNet dynamic routing, fused, WMMA-based (gfx1250 / CDNA5, wave32).
//
// Strategy (see analysis): never materialize the 302MB u_hat tensor.
// Each workgroup owns a (16-batch-row, single out-capsule j) slice and
// recomputes 16x16 u_hat tiles per input capsule i with chained
// V_WMMA_F32_16X16X4_F32 (K=8 -> 2 WMMAs). Routing logits are implicit:
//   b_t[i] = Vsum_t . u_hat[i,:]   with Vsum_t = v_0 + ... + v_{t-1}
// so each iteration is an online-softmax pass (running max/Z, flash style)
// plus a weighted-accumulation pass, all in ~19KB LDS. The 3 routing
// iterations are 3 launches of the same kernel; Vsum is carried in d_out
// (launch 0/1 write Vsum_{t+1}; launch 2 writes the final v). Zero d_ws use.

typedef float v2f __attribute__((ext_vector_type(2)));
typedef float v8f __attribute__((ext_vector_type(8)));

#define B_TOT    128
#define IN_CAPS  1152
#define OUT_CAPS 32
#define IN_D     8
#define OUT_D    16
#define NWAVE    8      // 256 threads = 8 wave32
#define NEG_BIG  (-3.0e38f)

__device__ __forceinline__ v8f wmma4(v2f a, v2f b, v8f c) {
  // 8-arg pattern: (neg_a, A, neg_b, B, c_mod, C, reuse_a, reuse_b)
  return __builtin_amdgcn_wmma_f32_16x16x4_f32(false, a, false, b,
                                               (short)0, c, false, false);
}

// One 16(batch) x 16(out_d) u_hat tile for input capsule i:
//   uh[bb,m] = sum_{n=0..7} u[b0+bb, i, n] * W[i, j, n, m]
// A layout (32b 16x4, ISA 7.12.2): lane<16 holds M=lane, K={0,1} in {x,y};
// lane>=16 holds M=lane-16, K={2,3}. B mirrors with K rows striped the same
// way (VGPR q: K=q+2h, N=lane&15). C/D: VGPR q holds row M=q+8h, N=lane&15.
__device__ __forceinline__ v8f uhat_tile(const float* __restrict__ uRow,
                                         const float* __restrict__ wTile) {
  v2f a1 = *(const v2f*)(uRow);        // n = 2h, 2h+1
  v2f a2 = *(const v2f*)(uRow + 4);    // n = 4+2h, 5+2h
  v2f b1, b2;
  b1.x = wTile[0];        // W(n=2h,   m)
  b1.y = wTile[OUT_D];    // W(n=2h+1, m)
  b2.x = wTile[4*OUT_D];  // W(n=4+2h, m)
  b2.y = wTile[5*OUT_D];  // W(n=5+2h, m)
  v8f c = {0.f,0.f,0.f,0.f,0.f,0.f,0.f,0.f};
  c = wmma4(a1, b1, c);
  c = wmma4(a2, b2, c);   // accumulate D->C (no A/B RAW hazard)
  return c;
}

__global__ __launch_bounds__(256) void caps_route_kernel(
    const float* __restrict__ u,    // [128][1152][8]
    const float* __restrict__ w,    // [1152][32][8][16]
    float* __restrict__ out,        // [128][32][16] : Vsum carrier / final v
    int t)                          // routing iteration 0..2
{
  __shared__ float tileS[NWAVE * 256];  // per-wave 16x16 uh scratch
  __shared__ float eS   [NWAVE * 16];   // per-wave softmax weights for one i
  __shared__ float redMx[NWAVE * 16];
  __shared__ float redZ [NWAVE * 16];
  __shared__ float mxS[16];
  __shared__ float izS[16];
  __shared__ float sredS[NWAVE * 256];  // cross-wave S reduction
  __shared__ float sfin[256];

  const int tid  = threadIdx.x;
  const int lane = tid & 31;
  const int wave = tid >> 5;
  const int mcol = lane & 15;          // N column / A row (M)
  const int h    = lane >> 4;          // lane half selects K pair
  const int b0   = blockIdx.x * 16;    // batch tile
  const int j    = blockIdx.y;         // out capsule

  const float* uBase = u + (size_t)(b0 + mcol) * IN_CAPS * IN_D + 2 * h;
  const float* wBase = w + (size_t)j * (IN_D * OUT_D) + (2 * h) * OUT_D + mcol;
  float* myTile = tileS + wave * 256;
  const int rowbase = 128 * h + mcol;  // (q+8h)*16+mcol = q*16+rowbase

  // ---- Vsum_t (prefix sum of routed v's), carried in d_out ----
  float vs[16];
#pragma unroll
  for (int m = 0; m < 16; ++m) vs[m] = 0.f;
  if (t > 0 && lane < 16) {
    const float4* vp =
        (const float4*)(out + ((size_t)(b0 + lane) * OUT_CAPS + j) * OUT_D);
    float4 q0 = vp[0], q1 = vp[1], q2 = vp[2], q3 = vp[3];
    vs[0]=q0.x; vs[1]=q0.y; vs[2]=q0.z; vs[3]=q0.w;
    vs[4]=q1.x; vs[5]=q1.y; vs[6]=q1.z; vs[7]=q1.w;
    vs[8]=q2.x; vs[9]=q2.y; vs[10]=q2.z; vs[11]=q2.w;
    vs[12]=q3.x; vs[13]=q3.y; vs[14]=q3.z; vs[15]=q3.w;
  }

  // ---- Pass 1 (t>0): logits d = Vsum.u_hat, online softmax max/Z ----
  if (t > 0) {
    float mxr = NEG_BIG, zr = 0.f;
    for (int i = wave; i < IN_CAPS; i += NWAVE) {
      v8f uh = uhat_tile(uBase + (size_t)i * IN_D,
                         wBase + (size_t)i * (OUT_CAPS * IN_D * OUT_D));
#pragma unroll
      for (int q = 0; q < 8; ++q) myTile[q * 16 + rowbase] = uh[q];
      if (lane < 16) {                       // lane = batch row bb
        const float* tr = myTile + lane * 16;
        float d = 0.f;
#pragma unroll
        for (int m = 0; m < 16; ++m) d += tr[m] * vs[m];
        float nm = fmaxf(mxr, d);            // flash-style running max/Z
        zr = zr * __expf(mxr - nm) + __expf(d - nm);
        mxr = nm;
      }
    }
    if (lane < 16) { redMx[wave * 16 + lane] = mxr; redZ[wave * 16 + lane] = zr; }
    __syncthreads();
    if (tid < 16) {
      float m = NEG_BIG;
      for (int wv = 0; wv < NWAVE; ++wv) m = fmaxf(m, redMx[wv * 16 + tid]);
      float z = 0.f;
      for (int wv = 0; wv < NWAVE; ++wv)
        z += redZ[wv * 16 + tid] * __expf(redMx[wv * 16 + tid] - m);
      mxS[tid] = m;
      izS[tid] = 1.0f / z;
    }
    __syncthreads();
  } else if (tid < 16) {
    izS[tid] = 1.0f / (float)IN_CAPS;        // softmax(0) = uniform
  }

  float mxreg = 0.f;
  if (t > 0 && lane < 16) mxreg = mxS[lane];

  // ---- Pass 2: S[bb,m] = sum_i exp(d_i - mx) * uh_i[bb,m] ----
  float Sacc[8];
#pragma unroll
  for (int q = 0; q < 8; ++q) Sacc[q] = 0.f;

  for (int i = wave; i < IN_CAPS; i += NWAVE) {
    v8f uh = uhat_tile(uBase + (size_t)i * IN_D,
                       wBase + (size_t)i * (OUT_CAPS * IN_D * OUT_D));
    if (t == 0) {
#pragma unroll
      for (int q = 0; q < 8; ++q) Sacc[q] += uh[q];
    } else {
#pragma unroll
      for (int q = 0; q < 8; ++q) myTile[q * 16 + rowbase] = uh[q];
      if (lane < 16) {
        const float* tr = myTile + lane * 16;
        float d = 0.f;
#pragma unroll
        for (int m = 0; m < 16; ++m) d += tr[m] * vs[m];
        eS[wave * 16 + lane] = __expf(d - mxreg);
      }
      const float* ep = eS + wave * 16 + 8 * h;  // e for bb = 8h+q
#pragma unroll
      for (int q = 0; q < 8; ++q) Sacc[q] += ep[q] * uh[q];
    }
  }

  // ---- Cross-wave S reduction, normalize, squash, write ----
#pragma unroll
  for (int q = 0; q < 8; ++q) sredS[wave * 256 + q * 16 + rowbase] = Sacc[q];
  __syncthreads();
  {
    float acc = 0.f;
    for (int wv = 0; wv < NWAVE; ++wv) acc += sredS[wv * 256 + tid];
    sfin[tid] = acc * izS[tid >> 4];           // tid = bb*16 + m
  }
  __syncthreads();
  if (tid < 16) {                              // tid = bb; has vs[] regs
    float sv[16], s2 = 0.f;
#pragma unroll
    for (int m = 0; m < 16; ++m) { sv[m] = sfin[tid * 16 + m]; s2 += sv[m] * sv[m]; }
    float coef = (s2 / (1.0f + s2)) / sqrtf(s2 + 1e-8f);   // squash
    float* op = out + ((size_t)(b0 + tid) * OUT_CAPS + j) * OUT_D;
#pragma unroll
    for (int m = 0; m < 16; ++m) {
      float v = coef * sv[m];
      op[m] = (t == 2) ? v : (vs[m] + v);      // carry Vsum_{t+1}, else final v
    }
  }
}

extern "C" void kernel_launch(void* const* d_in, const int* in_sizes, int n_in,
                              void* d_out, int out_size, void* d_ws, size_t ws_size,
                              hipStream_t stream) {
  (void)in_sizes; (void)n_in; (void)d_ws; (void)ws_size; (void)out_size;
  const float* u = (const float*)d_in[0];   // [128,1152,8]  f32
  const float* w = (const float*)d_in[1];   // [1152,32,8,16] f32
  // d_in[2] = iterations (device scalar); reference fixes it to 3.
  float* out = (float*)d_out;               // [128,32,16] f32

  dim3 grid(B_TOT / 16, OUT_CAPS);          // (8 batch tiles, 32 out caps)
  dim3 block(32 * NWAVE);                   // 8 wave32
  for (int t = 0; t < 3; ++t)
    caps_route_kernel<<<grid, block, 0, stream>>>(u, w, out, t);
}